// v_LPTC_7808250544862
// MI455X (gfx1250) — compile-verified
//
#include <hip/hip_runtime.h>

// CDNA5 / gfx1250: wave32, WMMA f32 16x16x4.
typedef float v2f __attribute__((ext_vector_type(2)));
typedef float v8f __attribute__((ext_vector_type(8)));

#define VOL 8192  // 16*8*8*8 lattice sites

// features_in : [4][VOL][4 spin][3 color]            complex64
// weights     : [4 fin][4 fout][9 path][VOL][4][4]   complex64
// U           : [4 mu][VOL][3][3]                    complex64
// out         : [4 fout][VOL][4 spin][3 color]       complex64
//
// Per site:  out[o,s,c] = sum_{i,p,j} W[i,o,p,site][s,j] * T[i,p,site][j,c]
// Realified as D(16x16) = A(16xK) * B(Kx16), K = (i,p,j,re/im) = 288:
//   A rows (M) = (color, re/im): [ Tr -Ti ; Ti Tr ] blocks, rows 6..15 = 0
//   B cols (N) = (o, s): one float2 (Wr,Wi) per lane per k-step, each weight
//                byte touched exactly once (stream, non-temporal)
__global__ __launch_bounds__(256, 2) void lptc_wmma_kernel(
    const float2* __restrict__ F,
    const float2* __restrict__ W,
    const float2* __restrict__ U,
    float2* __restrict__ OUT)
{
    // Pre-realified A matrix per wave: 8 rows x 288 K-floats (rows 6,7 zero).
    __shared__ float Aw[8][8][288];

    const int tid  = threadIdx.x;
    const int lane = tid & 31;
    const int wv   = tid >> 5;
    const int site = blockIdx.x * 8 + wv;   // one site per wave
    const int t = site >> 9, x = (site >> 6) & 7, y = (site >> 3) & 7, z = site & 7;

    // Periodic neighbors for each direction mu.
    int nplus[4], nminus[4];
    nplus [0] = (((t + 1) & 15) << 9) | (x << 6) | (y << 3) | z;
    nminus[0] = (((t + 15) & 15) << 9) | (x << 6) | (y << 3) | z;
    nplus [1] = (t << 9) | (((x + 1) & 7) << 6) | (y << 3) | z;
    nminus[1] = (t << 9) | (((x + 7) & 7) << 6) | (y << 3) | z;
    nplus [2] = (t << 9) | (x << 6) | (((y + 1) & 7) << 3) | z;
    nminus[2] = (t << 9) | (x << 6) | (((y + 7) & 7) << 3) | z;
    nplus [3] = (t << 9) | (x << 6) | (y << 3) | ((z + 1) & 7);
    nminus[3] = (t << 9) | (x << 6) | (y << 3) | ((z + 7) & 7);

    // ---- Phase 1: gauge transport + complex->real A expansion (tiny cost) ----
    // Zero rows 6,7 (read by the 10 unused A rows).
    for (int k2 = lane; k2 < 144; k2 += 32) {
        *(v2f*)&Aw[wv][6][k2 * 2] = (v2f){0.f, 0.f};
        *(v2f*)&Aw[wv][7][k2 * 2] = (v2f){0.f, 0.f};
    }
    // p = 0 : identity path.
    for (int cidx = lane; cidx < 48; cidx += 32) {
        int i = cidx / 12, r = cidx % 12;               // r = j*3 + c
        int j = r / 3, c = r % 3;
        float2 f = F[(i * VOL + site) * 12 + r];
        int ipj = i * 36 + j;
        *(v2f*)&Aw[wv][c * 2    ][ipj * 2] = (v2f){f.x, -f.y};
        *(v2f*)&Aw[wv][c * 2 + 1][ipj * 2] = (v2f){f.y,  f.x};
    }
    // p = 1..8 : one hop. odd p -> +mu: U_mu(x) f(x+mu); even -> U_mu^dag(x-mu) f(x-mu).
    for (int p = 1; p < 9; ++p) {
        const int  mu   = (p - 1) >> 1;
        const bool plus = (p & 1);
        const int  nbr  = plus ? nplus[mu] : nminus[mu];
        const float2* Ub = U + (mu * VOL + (plus ? site : nbr)) * 9;
        for (int cidx = lane; cidx < 48; cidx += 32) {
            int i = cidx / 12, j = (cidx % 12) / 3, a = cidx % 3;
            const float2* fb = F + (i * VOL + nbr) * 12 + j * 3;
            float ar = 0.f, ai = 0.f;
            #pragma unroll
            for (int b = 0; b < 3; ++b) {
                float2 uv = plus ? Ub[a * 3 + b] : Ub[b * 3 + a];
                float2 fv = fb[b];
                float uim = plus ? uv.y : -uv.y;        // conjugate for -mu hops
                ar += uv.x * fv.x - uim * fv.y;
                ai += uv.x * fv.y + uim * fv.x;
            }
            int ipj = (i * 9 + p) * 4 + j;
            *(v2f*)&Aw[wv][a * 2    ][ipj * 2] = (v2f){ar, -ai};
            *(v2f*)&Aw[wv][a * 2 + 1][ipj * 2] = (v2f){ai,  ar};
        }
    }
    __syncthreads();

    // ---- Phase 2: contraction, pure ds_load / NT global_load / v_wmma loop ----
    const int hk  = lane >> 4;          // K half: low lanes K0/K1, high lanes K2/K3
    const int col = lane & 15;          // B/D column
    const int oo  = col >> 2;           // output feature
    const int ss  = col & 3;            // output spin
    const int row = lane & 15;          // A row (M)
    const int rowSafe = (row < 6) ? row : 6;    // rows >=6 read the zero row

    // A fragment pointer: float2 index = 2m + hk, m = k-step.
    const v2f* ap = (const v2f*)&Aw[wv][rowSafe][0] + hk;
    // Weight float2 index: ((i*4+o)*9+p)*VOL*16 + site*16 + s*4 + j,  j = 2*h + hk.
    const float2* __restrict__ wl = W + oo * 1179648 + site * 16 + ss * 4 + hk;

    v8f acc0 = {0.f, 0.f, 0.f, 0.f, 0.f, 0.f, 0.f, 0.f};
    v8f acc1 = {0.f, 0.f, 0.f, 0.f, 0.f, 0.f, 0.f, 0.f};

    for (int i = 0; i < 4; ++i) {
        #pragma unroll
        for (int p = 0; p < 9; ++p) {
            const float2* wip = wl + i * 4718592 + p * 131072;
            const int aip = (i * 9 + p) * 4;
            v2f b0 = __builtin_nontemporal_load((const v2f*)(wip));
            v2f a0 = ap[aip];
            acc0 = __builtin_amdgcn_wmma_f32_16x16x4_f32(
                false, a0, false, b0, (short)0, acc0, false, false);
            v2f b1 = __builtin_nontemporal_load((const v2f*)(wip + 2));
            v2f a1 = ap[aip + 2];
            acc1 = __builtin_amdgcn_wmma_f32_16x16x4_f32(
                false, a1, false, b1, (short)0, acc1, false, false);
        }
    }
    v8f acc = acc0 + acc1;

    // ---- Phase 3: store. Lane n<16 owns column (o,s); rows 2c/2c+1 = re/im. ----
    if (lane < 16) {
        float2* op = OUT + (oo * VOL + site) * 12 + ss * 3;
        #pragma unroll
        for (int c = 0; c < 3; ++c)
            op[c] = make_float2(acc[2 * c], acc[2 * c + 1]);
    }
}

extern "C" void kernel_launch(void* const* d_in, const int* in_sizes, int n_in,
                              void* d_out, int out_size, void* d_ws, size_t ws_size,
                              hipStream_t stream) {
    (void)in_sizes; (void)n_in; (void)d_ws; (void)ws_size; (void)out_size;
    const float2* F = (const float2*)d_in[0];   // features_in (complex64 as float2)
    const float2* W = (const float2*)d_in[1];   // weights
    const float2* U = (const float2*)d_in[2];   // gauge links
    float2* O = (float2*)d_out;
    // 8192 sites, one wave per site, 8 waves (256 threads) per block.
    lptc_wmma_kernel<<<dim3(VOL / 8), dim3(256), 0, stream>>>(F, W, U, O);
}